// GraphEncoder_15126874817010
// MI455X (gfx1250) — compile-verified
//
#include <hip/hip_runtime.h>
#include <hip/hip_bf16.h>

typedef __attribute__((ext_vector_type(2))) float v2f;
typedef __attribute__((ext_vector_type(8))) float v8f;

#define D_FEAT 128

// ---------------------------------------------------------------------------
// Edge scatter-add: one wave per edge; lane l carries features [4l, 4l+3].
// h[dst] += x[src]  (h must already contain the self term = x).
// ---------------------------------------------------------------------------
__global__ __launch_bounds__(256) void edge_scatter(
    const float* __restrict__ X, float* __restrict__ H,
    const long long* __restrict__ src, const long long* __restrict__ dst,
    int nedges)
{
    int gwave = (int)((blockIdx.x * (unsigned)blockDim.x + threadIdx.x) >> 5);
    int lane  = threadIdx.x & 31;
    if (gwave >= nedges) return;
    long long s = src[gwave];
    long long t = dst[gwave];
    const float4 v = *(const float4*)(X + (size_t)s * D_FEAT + lane * 4);
    float* hp = H + (size_t)t * D_FEAT + lane * 4;
    atomicAdd(hp + 0, v.x);
    atomicAdd(hp + 1, v.y);
    atomicAdd(hp + 2, v.z);
    atomicAdd(hp + 3, v.w);
}

// ---------------------------------------------------------------------------
// Out[nrows x 128] = act(A[nrows x 128] @ W[128 x 128] + bias)
// Block: 256 threads = 8 waves. Block tile: 128 rows x 128 cols.
// Wave w owns rows [row0 + 16w, +16) and all 8 column tiles (8 x v8f acc).
// W staged transposed + XOR-swizzled in 64KB LDS -> conflict-free ds_load_b64.
// WMMA: V_WMMA_F32_16X16X4_F32, fp32 in / fp32 accumulate (matches reference).
// ---------------------------------------------------------------------------
__global__ __launch_bounds__(256) void gemm_bias_act(
    const float* __restrict__ A, const float* __restrict__ W,
    const float* __restrict__ bias, float* __restrict__ Out,
    int nrows, int do_relu)
{
    __shared__ float WT[D_FEAT * D_FEAT];   // 64 KB

    // Stage W^T with swizzle: WT[n*128 + (k ^ ((n&15)<<2))] = W[k*128 + n]
    for (int i = threadIdx.x; i < D_FEAT * D_FEAT; i += 256) {
        int k = i >> 7;
        int n = i & 127;
        WT[n * D_FEAT + (k ^ ((n & 15) << 2))] = W[i];
    }
    __syncthreads();

    int wave  = threadIdx.x >> 5;
    int lane  = threadIdx.x & 31;
    int rlane = lane & 15;      // row within A-tile / col within B-tile
    int khalf = lane >> 4;      // 0: K=0,1   1: K=2,3  (ISA 16x4 f32 A layout)

    int row0 = blockIdx.x * 128 + wave * 16;
    if (row0 >= nrows) return;  // wave-uniform: EXEC stays all-ones for WMMA

    v8f acc[8];
#pragma unroll
    for (int t = 0; t < 8; ++t) acc[t] = v8f{0.f,0.f,0.f,0.f,0.f,0.f,0.f,0.f};

    int arow = row0 + rlane;
    const float* aptr = A + (size_t)(arow < nrows ? arow : 0) * D_FEAT;
    int sw = rlane << 2;        // swizzle term (bits 2..5)

    for (int k = 0; k < D_FEAT; k += 4) {
        int kk = k + khalf * 2;                       // even -> 8B aligned
        v2f a = *(const v2f*)(aptr + kk);
        int base = rlane * D_FEAT + (kk ^ sw);
#pragma unroll
        for (int t = 0; t < 8; ++t) {
            v2f b = *(const v2f*)&WT[base + t * 16 * D_FEAT];
            acc[t] = __builtin_amdgcn_wmma_f32_16x16x4_f32(
                false, a, false, b, (short)0, acc[t], false, false);
        }
    }

    // Epilogue: C/D layout — VGPR r: lanes 0-15 -> M=r, lanes 16-31 -> M=r+8
#pragma unroll
    for (int t = 0; t < 8; ++t) {
        int col = t * 16 + rlane;
        float bv = bias[col];
#pragma unroll
        for (int r = 0; r < 8; ++r) {
            int orow = row0 + khalf * 8 + r;
            float v = acc[t][r] + bv;
            if (do_relu) v = fmaxf(v, 0.f);
            if (orow < nrows) Out[(size_t)orow * D_FEAT + col] = v;
        }
    }
}

// ---------------------------------------------------------------------------
// Readout helpers
// ---------------------------------------------------------------------------
__global__ void zero_kernel(float* __restrict__ p, int n)
{
    int i = blockIdx.x * blockDim.x + threadIdx.x;
    if (i < n) p[i] = 0.f;
}

__global__ __launch_bounds__(256) void pool_scatter(
    const float* __restrict__ X, const long long* __restrict__ batch,
    float* __restrict__ out, int nnodes)
{
    int idx  = blockIdx.x * blockDim.x + threadIdx.x;
    int node = idx >> 5;
    int c    = idx & 31;
    if (node >= nnodes) return;
    long long g = batch[node];
    const float4 v = *(const float4*)(X + (size_t)node * D_FEAT + c * 4);
    float* op = out + (size_t)g * D_FEAT + c * 4;
    atomicAdd(op + 0, v.x);
    atomicAdd(op + 1, v.y);
    atomicAdd(op + 2, v.z);
    atomicAdd(op + 3, v.w);
}

// ---------------------------------------------------------------------------
// Launch
// ---------------------------------------------------------------------------
extern "C" void kernel_launch(void* const* d_in, const int* in_sizes, int n_in,
                              void* d_out, int out_size, void* d_ws, size_t ws_size,
                              hipStream_t stream)
{
    const float*     x0   = (const float*)d_in[0];
    const float*     W1   = (const float*)d_in[1];
    const float*     b1   = (const float*)d_in[2];
    const float*     W2   = (const float*)d_in[3];
    const float*     b2   = (const float*)d_in[4];
    const long long* eidx = (const long long*)d_in[5];
    const long long* batch= (const long long*)d_in[6];

    const int nnodes  = in_sizes[0] / D_FEAT;
    const int nedges  = in_sizes[5] / 2;
    const int nlayers = in_sizes[2] / D_FEAT;
    const int ngraphs = out_size / D_FEAT;

    const long long* src = eidx;
    const long long* dst = eidx + nedges;

    const size_t feat_elems = (size_t)nnodes * D_FEAT;
    float* bufA = (float*)d_ws;              // h = x + agg
    float* bufB = bufA + feat_elems;         // relu(h @ W1 + b1)
    float* bufC = bufB + feat_elems;         // x_next

    const int scat_blocks = (nedges + 7) / 8;            // 1 wave / edge
    const int gemm_blocks = (nnodes + 127) / 128;

    for (int l = 0; l < nlayers; ++l) {
        const float* Xi = (l == 0) ? x0 : bufC;
        // self term (eps = 0): h = x
        hipMemcpyAsync(bufA, Xi, feat_elems * sizeof(float),
                       hipMemcpyDeviceToDevice, stream);
        // h += sum of neighbor features
        edge_scatter<<<scat_blocks, 256, 0, stream>>>(Xi, bufA, src, dst, nedges);
        // t = relu(h @ W1 + b1)
        gemm_bias_act<<<gemm_blocks, 256, 0, stream>>>(
            bufA, W1 + (size_t)l * D_FEAT * D_FEAT, b1 + (size_t)l * D_FEAT,
            bufB, nnodes, 1);
        // x = t @ W2 + b2
        gemm_bias_act<<<gemm_blocks, 256, 0, stream>>>(
            bufB, W2 + (size_t)l * D_FEAT * D_FEAT, b2 + (size_t)l * D_FEAT,
            bufC, nnodes, 0);
    }

    // sum-pool readout
    int nout = ngraphs * D_FEAT;
    zero_kernel<<<(nout + 255) / 256, 256, 0, stream>>>((float*)d_out, nout);
    int pool_threads = nnodes * 32;
    pool_scatter<<<(pool_threads + 255) / 256, 256, 0, stream>>>(
        bufC, batch, (float*)d_out, nnodes);
}